// CPN_91053306675223
// MI455X (gfx1250) — compile-verified
//
#include <hip/hip_runtime.h>
#include <math.h>

#define Bv 16
#define Nv 2048
#define NDENSE 4096
#define LATENTv 1024
#define Pv 8
#define NCOARSE 1024
#define NPPv 128
#define Mv 3072
#define LDv 1026
#define ALPHAv 1.5f
#define BIGF 3.0e38f
#define KT 16   // K-chunk (rows) staged per TDM transfer
#define NT 32   // N-width of the TDM-staged output tile

typedef __attribute__((ext_vector_type(2))) float v2f;
typedef __attribute__((ext_vector_type(8))) float v8f;
typedef __attribute__((ext_vector_type(4))) unsigned int v4u;
typedef __attribute__((ext_vector_type(8))) int v8i;
typedef __attribute__((ext_vector_type(4))) int v4i;

#if __has_builtin(__builtin_amdgcn_tensor_load_to_lds)
#define HAVE_TDM 1
#else
#define HAVE_TDM 0
#endif

static __device__ __forceinline__ float actf(float v, int act) {
  if (act == 1) return v > 0.f ? v : 0.f;               // relu
  if (act == 2) return v > 0.f ? v : 0.1f * v;          // leaky 0.1
  if (act == 3) return v > 0.f ? v : (__expf(v) - 1.f); // elu
  return v;
}

#if HAVE_TDM
// Issue a TDM load of a [rows x cols] f32 tile (row stride ldx elements) to LDS.
// D# group0: count=1 | lds_addr | global_addr | type=2.
// D# group1: data_size=4B, tensor dims == tile dims (no OOB clipping needed),
//            tensor_dim0_stride = ldx.
static __device__ __forceinline__ void tdm_load_tile(void* ldsDst, const float* gsrc,
                                                     int ldx, int rows, int cols)
{
  const unsigned lds = (unsigned)(unsigned long long)ldsDst;  // low 32 bits = LDS offset
  const unsigned long long ga = (unsigned long long)gsrc;
  v4u g0;
  g0.x = 1u;                                                  // count=1 (valid user D#)
  g0.y = lds;                                                 // lds_addr
  g0.z = (unsigned)(ga & 0xFFFFFFFFull);                      // global_addr[31:0]
  g0.w = (unsigned)((ga >> 32) & 0x01FFFFFFull) | (2u << 30); // addr[56:32] | type=2
  v8i g1;
  g1[0] = 0x00020000;                                         // wg_mask=0, data_size=2 (4B)
  g1[1] = (int)(((unsigned)cols & 0xFFFFu) << 16);            // atomic_addr=0 | tensor_dim0.lo
  g1[2] = (int)(((unsigned)rows & 0xFFFFu) << 16);            // tensor_dim0.hi=0 | tensor_dim1.lo
  g1[3] = (int)(((unsigned)cols & 0xFFFFu) << 16);            // tensor_dim1.hi=0 | tile_dim0
  g1[4] = (int)((unsigned)rows & 0xFFFFu);                    // tile_dim1 | tile_dim2=0
  g1[5] = ldx;                                                // tensor_dim0_stride[31:0]
  g1[6] = 0;                                                  // stride hi | dim1_stride lo
  g1[7] = 0;
  v4i g2 = {0, 0, 0, 0};
  v4i g3 = {0, 0, 0, 0};
#if defined(__clang_major__) && __clang_major__ >= 23
  v8i g4 = {0, 0, 0, 0, 0, 0, 0, 0};
  __builtin_amdgcn_tensor_load_to_lds(g0, g1, g2, g3, g4, 0);
#else
  __builtin_amdgcn_tensor_load_to_lds(g0, g1, g2, g3, 0);
#endif
}
#endif

// ---------------------------------------------------------------------------
// Generic WMMA f32 GEMM (handles ragged O/I/N): one wave per 16x16 tile.
// Interior tiles run a branchless K-main loop (packed v2f A loads when I is
// even); edge tiles use clamped-address loads x 0/1 masks - no exec-mask
// churn anywhere in the K loop.
// ---------------------------------------------------------------------------
__global__ void k_wmma_gemm(const float* __restrict__ W, long long wStrideZ, int wZdiv,
                            const float* __restrict__ X, long long xStrideZ, int ldx,
                            float* __restrict__ Cp, long long cStrideZ, int ldc,
                            const float* __restrict__ bias,
                            int O, int I, int Ncols, int act)
{
  const int lane = threadIdx.x & 31;
  const int m    = lane & 15;
  const int kb   = (lane >> 4) << 1;
  const int o0   = blockIdx.y << 4, n0 = blockIdx.x << 4;
  const int z    = blockIdx.z;
  const float* Wz = W + (long long)(z / wZdiv) * wStrideZ;
  const float* Xz = X + (long long)z * xStrideZ;
  float*       Cz = Cp + (long long)z * cStrideZ;
  const int ra = o0 + m;
  const int cb = n0 + m;
  const int Imain = I & ~3;
  const bool interior = (o0 + 16 <= O) && (n0 + 16 <= Ncols);
  v8f acc = {};
  if (interior && !(I & 1)) {
    const float* Wr = Wz + (long long)ra * I + kb;       // 8B-aligned (I even)
    const float* Xc = Xz + (long long)kb * ldx + cb;
    for (int k0 = 0; k0 < Imain; k0 += 4) {
      const v2f a = *(const v2f*)(Wr + k0);
      v2f b;
      b.x = Xc[(long long)k0 * ldx];
      b.y = Xc[(long long)(k0 + 1) * ldx];
      acc = __builtin_amdgcn_wmma_f32_16x16x4_f32(false, a, false, b, (short)0, acc, false, false);
    }
  } else if (interior) {
    const float* Wr = Wz + (long long)ra * I + kb;
    const float* Xc = Xz + (long long)kb * ldx + cb;
    for (int k0 = 0; k0 < Imain; k0 += 4) {
      v2f a, b;
      a.x = Wr[k0];
      a.y = Wr[k0 + 1];
      b.x = Xc[(long long)k0 * ldx];
      b.y = Xc[(long long)(k0 + 1) * ldx];
      acc = __builtin_amdgcn_wmma_f32_16x16x4_f32(false, a, false, b, (short)0, acc, false, false);
    }
  } else {
    // branchless edge path: clamp addresses into bounds, mask contributions
    const int   rac = (ra < O) ? ra : (O - 1);
    const float am  = (ra < O) ? 1.f : 0.f;
    const int   cbc = (cb < Ncols) ? cb : (Ncols - 1);
    const float bm  = (cb < Ncols) ? 1.f : 0.f;
    const float* Wr = Wz + (long long)rac * I + kb;
    const float* Xc = Xz + (long long)kb * ldx + cbc;
    for (int k0 = 0; k0 < Imain; k0 += 4) {
      v2f a, b;
      a.x = Wr[k0] * am;
      a.y = Wr[k0 + 1] * am;
      b.x = Xc[(long long)k0 * ldx] * bm;
      b.y = Xc[(long long)(k0 + 1) * ldx] * bm;
      acc = __builtin_amdgcn_wmma_f32_16x16x4_f32(false, a, false, b, (short)0, acc, false, false);
    }
  }
  if (Imain < I) {  // K remainder: clamp K too, mask out-of-range K lanes
    const int ka  = Imain + kb;
    const int rac = (ra < O) ? ra : (O - 1);
    const int cbc = (cb < Ncols) ? cb : (Ncols - 1);
    const float am = ((ra < O) ? 1.f : 0.f);
    const float bm = ((cb < Ncols) ? 1.f : 0.f);
    const int k0c = (ka < I) ? ka : (I - 1);
    const int k1c = (ka + 1 < I) ? (ka + 1) : (I - 1);
    const float km0 = (ka < I) ? 1.f : 0.f;
    const float km1 = (ka + 1 < I) ? 1.f : 0.f;
    v2f a, b;
    a.x = Wz[(long long)rac * I + k0c] * am * km0;
    a.y = Wz[(long long)rac * I + k1c] * am * km1;
    b.x = Xz[(long long)k0c * ldx + cbc] * bm;
    b.y = Xz[(long long)k1c * ldx + cbc] * bm;
    acc = __builtin_amdgcn_wmma_f32_16x16x4_f32(false, a, false, b, (short)0, acc, false, false);
  }
  const int rowb = o0 + ((lane >> 4) << 3);
#pragma unroll
  for (int rr = 0; rr < 8; ++rr) {
    const int o = rowb + rr, n = n0 + m;
    if (o < O && n < Ncols) {
      float v = acc[rr];
      if (bias) v += bias[o];
      Cz[(long long)o * ldc + n] = actf(v, act);
    }
  }
}

// ---------------------------------------------------------------------------
// Regular WMMA f32 GEMM (O%16==0, N%32==0, I%KT==0): 16x32 output per wave.
// B-tile (KT x 32 f32) double-buffered in LDS via the Tensor Data Mover; one
// A fragment (packed v2f, global_load_b64) feeds two WMMAs per K-step, synced
// with s_wait_tensorcnt. DMA of chunk ch+1 overlaps the 8 WMMAs of chunk ch.
// ---------------------------------------------------------------------------
__global__ void k_wmma_gemm_tdm(const float* __restrict__ W, long long wStrideZ, int wZdiv,
                                const float* __restrict__ X, long long xStrideZ, int ldx,
                                float* __restrict__ Cp, long long cStrideZ, int ldc,
                                const float* __restrict__ bias,
                                int O, int I, int Ncols, int act)
{
  __shared__ float tile[2][KT * NT];
  const int lane = threadIdx.x & 31;
  const int m    = lane & 15;
  const int kb   = (lane >> 4) << 1;
  const int o0   = blockIdx.y << 4, n0 = blockIdx.x << 5;
  const int z    = blockIdx.z;
  const float* Wz = W + (long long)(z / wZdiv) * wStrideZ;
  const float* Xz = X + (long long)z * xStrideZ;
  float*       Cz = Cp + (long long)z * cStrideZ;
  const int ra = o0 + m;
  const int nch = I / KT;
#if HAVE_TDM
  tdm_load_tile(&tile[0][0], Xz + n0, ldx, KT, NT);
#endif
  v8f acc0 = {}, acc1 = {};
  for (int ch = 0; ch < nch; ++ch) {
#if HAVE_TDM
    if (ch + 1 < nch) {
      tdm_load_tile(&tile[(ch + 1) & 1][0], Xz + (long long)(ch + 1) * KT * ldx + n0, ldx, KT, NT);
      __builtin_amdgcn_s_wait_tensorcnt(1);
    } else {
      __builtin_amdgcn_s_wait_tensorcnt(0);
    }
#else
    for (int t = lane; t < KT * NT; t += 32)
      tile[ch & 1][t] = Xz[((long long)ch * KT + (t / NT)) * ldx + n0 + (t % NT)];
#endif
    const float* bt = tile[ch & 1];
    const float* Wr = Wz + (long long)ra * I + ch * KT + kb;   // 8B-aligned (I%16==0)
#pragma unroll
    for (int k4 = 0; k4 < KT; k4 += 4) {
      const v2f a = *(const v2f*)(Wr + k4);
      v2f b0, b1;
      b0.x = bt[(k4 + kb) * NT + m];
      b0.y = bt[(k4 + kb + 1) * NT + m];
      b1.x = bt[(k4 + kb) * NT + 16 + m];
      b1.y = bt[(k4 + kb + 1) * NT + 16 + m];
      acc0 = __builtin_amdgcn_wmma_f32_16x16x4_f32(false, a, false, b0, (short)0, acc0, false, false);
      acc1 = __builtin_amdgcn_wmma_f32_16x16x4_f32(false, a, false, b1, (short)0, acc1, false, false);
    }
  }
  const int rowb = o0 + ((lane >> 4) << 3);
#pragma unroll
  for (int rr = 0; rr < 8; ++rr) {
    const int o = rowb + rr;
    float v0 = acc0[rr], v1 = acc1[rr];
    if (bias) { v0 += bias[o]; v1 += bias[o]; }
    Cz[(long long)o * ldc + n0 + m]      = actf(v0, act);
    Cz[(long long)o * ldc + n0 + 16 + m] = actf(v1, act);
  }
}

// ---------------------------------------------------------------------------
// WMMA 3x3x3 voxel conv, padding 1, leaky-relu(0.1) epilogue. Boundary taps use
// clamped-address loads multiplied by a 0/1 mask (no exec-mask churn). O and
// r^3 are always multiples of 16 here, so only the I%4 remainder is guarded.
// ---------------------------------------------------------------------------
__global__ void k_wmma_voxconv(const float* __restrict__ Wv, const float* __restrict__ G,
                               float* __restrict__ Vg, int O, int I, int r, int rs)
{
  const int r3   = r * r * r;
  const int lane = threadIdx.x & 31;
  const int m    = lane & 15;
  const int kb   = (lane >> 4) << 1;
  const int o0   = blockIdx.y << 4, n0 = blockIdx.x << 4;
  const int z    = blockIdx.z;
  const float* Gz = G + (long long)z * I * r3;
  const int v  = n0 + m;
  const int d2 = v & (r - 1), d1 = (v >> rs) & (r - 1), d0 = v >> (2 * rs);
  const int ra = o0 + m;
  const int Imain = I & ~3;
  v8f acc = {};
  for (int t = 0; t < 27; ++t) {
    const int t0 = t / 9, t1 = (t / 3) % 3, t2 = t % 3;
    const int e0 = d0 + t0 - 1, e1 = d1 + t1 - 1, e2 = d2 + t2 - 1;
    const bool inb = (e0 >= 0) && (e0 < r) && (e1 >= 0) && (e1 < r) && (e2 >= 0) && (e2 < r);
    const float msk = inb ? 1.f : 0.f;
    const int nbc = inb ? (e0 * r + e1) * r + e2 : 0;  // clamped neighbor
    const float* Wr = Wv + ((long long)ra * I + kb) * 27 + t;
    const float* Gc = Gz + (long long)kb * r3 + nbc;
    for (int k0 = 0; k0 < Imain; k0 += 4) {
      v2f a, b;
      a.x = Wr[(long long)k0 * 27];
      a.y = Wr[(long long)(k0 + 1) * 27];
      b.x = Gc[(long long)k0 * r3] * msk;
      b.y = Gc[(long long)(k0 + 1) * r3] * msk;
      acc = __builtin_amdgcn_wmma_f32_16x16x4_f32(false, a, false, b, (short)0, acc, false, false);
    }
    if (Imain < I) {
      const int ka = Imain + kb;
      const int k0c = (ka < I) ? ka : (I - 1);
      const int k1c = (ka + 1 < I) ? (ka + 1) : (I - 1);
      const float km0 = (ka < I) ? 1.f : 0.f;
      const float km1 = (ka + 1 < I) ? 1.f : 0.f;
      v2f a, b;
      a.x = Wv[((long long)ra * I + k0c) * 27 + t] * km0;
      a.y = Wv[((long long)ra * I + k1c) * 27 + t] * km1;
      b.x = Gz[(long long)k0c * r3 + nbc] * msk;
      b.y = Gz[(long long)k1c * r3 + nbc] * msk;
      acc = __builtin_amdgcn_wmma_f32_16x16x4_f32(false, a, false, b, (short)0, acc, false, false);
    }
  }
  float* Vz = Vg + (long long)z * O * r3;
  const int rowb = o0 + ((lane >> 4) << 3);
#pragma unroll
  for (int rr = 0; rr < 8; ++rr) {
    const float val = acc[rr];
    Vz[(long long)(rowb + rr) * r3 + n0 + m] = val > 0.f ? val : 0.1f * val;
  }
}

// ------------------------------- support kernels ----------------------------
__global__ void k_transpose6(const float* __restrict__ in, float* __restrict__ out, int n)
{
  long long id = (long long)blockIdx.x * 256 + threadIdx.x;
  if (id >= (long long)Bv * 6 * n) return;
  const int i = (int)(id % n);
  long long t = id / n;
  const int c = (int)(t % 6);
  const int b = (int)(t / 6);
  out[id] = in[((long long)b * n + i) * 6 + c];
}

__global__ void k_stats(const float* __restrict__ xt, long long czs, int n, float* __restrict__ stats)
{
  const int b = blockIdx.x, tid = threadIdx.x;  // 256 threads
  __shared__ float s0[256], s1[256], s2[256];
  __shared__ float mean0, mean1, mean2;
  const float* cb = xt + (long long)b * czs;
  float a0 = 0.f, a1 = 0.f, a2 = 0.f;
  for (int i = tid; i < n; i += 256) { a0 += cb[i]; a1 += cb[n + i]; a2 += cb[2 * n + i]; }
  s0[tid] = a0; s1[tid] = a1; s2[tid] = a2; __syncthreads();
  for (int w = 128; w > 0; w >>= 1) {
    if (tid < w) { s0[tid] += s0[tid + w]; s1[tid] += s1[tid + w]; s2[tid] += s2[tid + w]; }
    __syncthreads();
  }
  if (tid == 0) { mean0 = s0[0] / n; mean1 = s1[0] / n; mean2 = s2[0] / n; }
  __syncthreads();
  float mx = 0.f;
  for (int i = tid; i < n; i += 256) {
    const float dx = cb[i] - mean0, dy = cb[n + i] - mean1, dz = cb[2 * n + i] - mean2;
    mx = fmaxf(mx, sqrtf(dx * dx + dy * dy + dz * dz));
  }
  s0[tid] = mx; __syncthreads();
  for (int w = 128; w > 0; w >>= 1) { if (tid < w) s0[tid] = fmaxf(s0[tid], s0[tid + w]); __syncthreads(); }
  if (tid == 0) { stats[b * 4 + 0] = mean0; stats[b * 4 + 1] = mean1; stats[b * 4 + 2] = mean2; stats[b * 4 + 3] = s0[0]; }
}

__global__ void k_flat(const float* __restrict__ xt, long long czs, int n,
                       const float* __restrict__ stats, int r, int* __restrict__ flat)
{
  long long id = (long long)blockIdx.x * 256 + threadIdx.x;
  if (id >= (long long)Bv * n) return;
  const int b = (int)(id / n), i = (int)(id % n);
  const float* cb = xt + (long long)b * czs;
  const float inv = 1.f / (2.f * stats[b * 4 + 3] + 1e-6f);
  int vv[3];
#pragma unroll
  for (int c = 0; c < 3; ++c) {
    const float cc = cb[(long long)c * n + i] - stats[b * 4 + c];
    float f = floorf((cc * inv + 0.5f) * (float)r);
    f = fminf(fmaxf(f, 0.f), (float)(r - 1));
    vv[c] = (int)f;
  }
  flat[(long long)b * n + i] = (vv[0] * r + vv[1]) * r + vv[2];
}

__global__ void k_scatter(const float* __restrict__ F, long long fzs, int ldf, int C,
                          const int* __restrict__ flat, int n, int r3,
                          float* __restrict__ sums, float* __restrict__ cnt)
{
  long long id = (long long)blockIdx.x * 256 + threadIdx.x;
  if (id >= (long long)Bv * n) return;
  const int b = (int)(id / n), i = (int)(id % n);
  const int fl = flat[(long long)b * n + i];
  atomicAdd(&cnt[(long long)b * r3 + fl], 1.f);
  const float* Fb = F + (long long)b * fzs;
  float* Sb = sums + (long long)b * C * r3;
  for (int c = 0; c < C; ++c) atomicAdd(&Sb[(long long)c * r3 + fl], Fb[(long long)c * ldf + i]);
}

__global__ void k_avg(float* __restrict__ sums, const float* __restrict__ cnt, int C, int r3)
{
  long long id = (long long)blockIdx.x * 256 + threadIdx.x;
  if (id >= (long long)Bv * C * r3) return;
  const int vox = (int)(id % r3);
  const int b = (int)(id / ((long long)C * r3));
  sums[id] /= fmaxf(cnt[(long long)b * r3 + vox], 1.f);
}

__global__ void k_gather_add(const float* __restrict__ pt, const float* __restrict__ Vg,
                             const int* __restrict__ flat, float* __restrict__ outp,
                             long long ozs, int ldo, int O, int n, int r3)
{
  long long id = (long long)blockIdx.x * 256 + threadIdx.x;
  if (id >= (long long)Bv * O * n) return;
  const int i = (int)(id % n);
  long long t = id / n;
  const int o = (int)(t % O);
  const int b = (int)(t / O);
  const float v = pt[((long long)b * O + o) * (long long)n + i] +
                  Vg[((long long)b * O + o) * (long long)r3 + flat[(long long)b * n + i]];
  outp[(long long)b * ozs + (long long)o * ldo + i] = v;
}

__global__ void k_maxpool(const float* __restrict__ feats, float* __restrict__ fm_t)
{
  const int id = blockIdx.x * 256 + threadIdx.x;
  if (id >= LATENTv * Bv) return;
  const int c = id / Bv, b = id % Bv;
  const float* row = feats + ((long long)b * LATENTv + c) * Nv;
  float mx = -BIGF;
  for (int i = 0; i < Nv; ++i) mx = fmaxf(mx, row[i]);
  fm_t[(long long)c * Bv + b] = mx;
}

__global__ void k_build_yin(const float* __restrict__ rg, const float* __restrict__ feat_t,
                            float* __restrict__ yin)
{
  long long id = (long long)blockIdx.x * 256 + threadIdx.x;
  if (id >= (long long)Pv * Bv * LDv * NPPv) return;
  const int nn = (int)(id % NPPv);
  long long t = id / NPPv;
  const int i = (int)(t % LDv); t /= LDv;
  const int b = (int)(t % Bv);
  const int p = (int)(t / Bv);
  float v;
  if (i < 2) v = rg[(((long long)p * Bv + b) * 2 + i) * NPPv + nn];
  else       v = feat_t[(long long)(i - 2) * Bv + b];
  yin[id] = v;
}

__global__ void k_coarse(const float* __restrict__ y4, float* __restrict__ coarse)
{
  long long id = (long long)blockIdx.x * 256 + threadIdx.x;
  if (id >= (long long)Bv * NCOARSE * 6) return;
  const int c = (int)(id % 6);
  long long t = id / 6;
  const int pc = (int)(t % NCOARSE);
  const int b = (int)(t / NCOARSE);
  const int p = pc / NPPv, nn = pc % NPPv;
  coarse[id] = y4[(((long long)p * Bv + b) * 6 + c) * NPPv + nn];
}

// MST (Prim) per group of NPP points; accumulates exp_loss and mean_mst.
__global__ void k_mst(const float* __restrict__ coarse, float* __restrict__ meanmst,
                      float* __restrict__ expLoss)
{
  const int g = blockIdx.x;               // B*P groups
  const int b = g / Pv, p = g % Pv;
  const int tid = threadIdx.x;            // NPP threads
  __shared__ float px[NPPv], py[NPPv], pz[NPPv], mind[NPPv], lens[NPPv];
  __shared__ int idxs[NPPv];
  __shared__ unsigned char intree[NPPv];
  __shared__ float red[NPPv]; __shared__ int redi[NPPv];
  const float* base = coarse + ((long long)b * NCOARSE + (long long)p * NPPv) * 6;
  px[tid] = base[tid * 6 + 0]; py[tid] = base[tid * 6 + 1]; pz[tid] = base[tid * 6 + 2];
  intree[tid] = (tid == 0);
  __syncthreads();
  {
    const float dx = px[tid] - px[0], dy = py[tid] - py[0], dz = pz[tid] - pz[0];
    mind[tid] = sqrtf(dx * dx + dy * dy + dz * dz);
  }
  __syncthreads();
  for (int s = 0; s < NPPv - 1; ++s) {
    red[tid] = intree[tid] ? BIGF : mind[tid];
    redi[tid] = tid;
    __syncthreads();
    for (int w = NPPv / 2; w > 0; w >>= 1) {
      if (tid < w) {
        if (red[tid + w] < red[tid] || (red[tid + w] == red[tid] && redi[tid + w] < redi[tid])) {
          red[tid] = red[tid + w]; redi[tid] = redi[tid + w];
        }
      }
      __syncthreads();
    }
    const int j = redi[0];
    if (tid == 0) { lens[s] = red[0]; idxs[s] = j; intree[j] = 1; }
    __syncthreads();
    const float dx = px[tid] - px[j], dy = py[tid] - py[j], dz = pz[tid] - pz[j];
    mind[tid] = fminf(mind[tid], sqrtf(dx * dx + dy * dy + dz * dz));
    __syncthreads();
  }
  red[tid] = (tid < NPPv - 1) ? lens[tid] : 0.f;
  __syncthreads();
  for (int w = NPPv / 2; w > 0; w >>= 1) { if (tid < w) red[tid] += red[tid + w]; __syncthreads(); }
  const float meanl = red[0] / (float)(NPPv - 1);
  const float pen = (tid < NPPv - 1 && lens[tid] > ALPHAv * meanl) ? lens[tid] : 0.f;
  __syncthreads();
  red[tid] = pen; __syncthreads();
  for (int w = NPPv / 2; w > 0; w >>= 1) { if (tid < w) red[tid] += red[tid + w]; __syncthreads(); }
  if (tid == 0) {
    atomicAdd(&meanmst[b], meanl / (float)Pv);
    atomicAdd(expLoss, red[0] / (float)(Bv * NCOARSE));
  }
}

// Farthest-point sampling over coarse (1024 pts / batch).
__global__ void k_fps(const float* __restrict__ coarse, int* __restrict__ idx)
{
  const int b = blockIdx.x, tid = threadIdx.x;  // 256 threads
  __shared__ float px[NCOARSE], py[NCOARSE], pz[NCOARSE], mind[NCOARSE];
  __shared__ float red[256]; __shared__ int redi[256];
  __shared__ float pj0, pj1, pj2;
  for (int i = tid; i < NCOARSE; i += 256) {
    const float* q = coarse + ((long long)b * NCOARSE + i) * 6;
    px[i] = q[0]; py[i] = q[1]; pz[i] = q[2];
  }
  __syncthreads();
  for (int i = tid; i < NCOARSE; i += 256) {
    const float dx = px[i] - px[0], dy = py[i] - py[0], dz = pz[i] - pz[0];
    mind[i] = dx * dx + dy * dy + dz * dz;
  }
  if (tid == 0) idx[(long long)b * NCOARSE] = 0;
  __syncthreads();
  for (int s = 1; s < NCOARSE; ++s) {
    float best = -1.f; int bi = 0;
    for (int i = tid; i < NCOARSE; i += 256) { const float v = mind[i]; if (v > best) { best = v; bi = i; } }
    red[tid] = best; redi[tid] = bi; __syncthreads();
    for (int w = 128; w > 0; w >>= 1) {
      if (tid < w) {
        if (red[tid + w] > red[tid] || (red[tid + w] == red[tid] && redi[tid + w] < redi[tid])) {
          red[tid] = red[tid + w]; redi[tid] = redi[tid + w];
        }
      }
      __syncthreads();
    }
    const int j = redi[0];
    if (tid == 0) { idx[(long long)b * NCOARSE + s] = j; pj0 = px[j]; pj1 = py[j]; pj2 = pz[j]; }
    __syncthreads();
    for (int i = tid; i < NCOARSE; i += 256) {
      const float dx = px[i] - pj0, dy = py[i] - pj1, dz = pz[i] - pj2;
      const float d2 = dx * dx + dy * dy + dz * dz;
      if (d2 < mind[i]) mind[i] = d2;
    }
    __syncthreads();
  }
}

__global__ void k_dense_gather(const float* __restrict__ coarse, const int* __restrict__ idx,
                               float* __restrict__ dense)
{
  long long id = (long long)blockIdx.x * 256 + threadIdx.x;
  if (id >= (long long)Bv * NDENSE * 6) return;
  const int c = (int)(id % 6);
  long long t = id / 6;
  const int j = (int)(t % NDENSE);
  const int b = (int)(t / NDENSE);
  dense[id] = (j < NCOARSE)
      ? coarse[((long long)b * NCOARSE + idx[(long long)b * NCOARSE + j]) * 6 + c]
      : 0.f;
}

__global__ void k_dense0raw(const float* __restrict__ xyz, const float* __restrict__ coarse,
                            float* __restrict__ d0)
{
  long long id = (long long)blockIdx.x * 256 + threadIdx.x;
  if (id >= (long long)Bv * Mv * 6) return;
  const int c = (int)(id % 6);
  long long t = id / 6;
  const int j = (int)(t % Mv);
  const int b = (int)(t / Mv);
  d0[id] = (j < Nv) ? xyz[((long long)b * Nv + j) * 6 + c]
                    : coarse[((long long)b * NCOARSE + (j - Nv)) * 6 + c];
}

// Sequential minimum-density ordering (3072 serial steps, LDS-resident state).
__global__ void k_mds(const float* __restrict__ d0raw, const float* __restrict__ meanmst,
                      int* __restrict__ order)
{
  const int b = blockIdx.x, tid = threadIdx.x;  // 256 threads
  __shared__ float sx[Mv], sy[Mv], sz[Mv], dens[Mv];
  __shared__ unsigned char sel[Mv];
  __shared__ float red[256]; __shared__ int redi[256];
  __shared__ float pj0, pj1, pj2;
  for (int i = tid; i < Mv; i += 256) {
    const float* q = d0raw + ((long long)b * Mv + i) * 6;
    sx[i] = q[0]; sy[i] = q[1]; sz[i] = q[2]; dens[i] = 0.f; sel[i] = 0;
  }
  const float sg = meanmst[b];
  const float s2 = 2.f * sg * sg + 1e-12f;
  __syncthreads();
  for (int s = 0; s < Mv; ++s) {
    float best = BIGF; int bi = 0;
    for (int i = tid; i < Mv; i += 256) { const float v = sel[i] ? BIGF : dens[i]; if (v < best) { best = v; bi = i; } }
    red[tid] = best; redi[tid] = bi; __syncthreads();
    for (int w = 128; w > 0; w >>= 1) {
      if (tid < w) {
        if (red[tid + w] < red[tid] || (red[tid + w] == red[tid] && redi[tid + w] < redi[tid])) {
          red[tid] = red[tid + w]; redi[tid] = redi[tid + w];
        }
      }
      __syncthreads();
    }
    const int j = redi[0];
    if (tid == 0) { sel[j] = 1; order[(long long)b * Mv + s] = j; pj0 = sx[j]; pj1 = sy[j]; pj2 = sz[j]; }
    __syncthreads();
    const float ax = pj0, ay = pj1, az = pj2;
    for (int i = tid; i < Mv; i += 256) {
      const float dx = sx[i] - ax, dy = sy[i] - ay, dz = sz[i] - az;
      dens[i] += __expf(-(dx * dx + dy * dy + dz * dz) / s2);
    }
    __syncthreads();
  }
}

__global__ void k_reorder(const float* __restrict__ d0, const int* __restrict__ order,
                          float* __restrict__ outp)
{
  long long id = (long long)blockIdx.x * 256 + threadIdx.x;
  if (id >= (long long)Bv * Mv * 6) return;
  const int c = (int)(id % 6);
  long long t = id / 6;
  const int j = (int)(t % Mv);
  const int b = (int)(t / Mv);
  outp[id] = d0[((long long)b * Mv + order[(long long)b * Mv + j]) * 6 + c];
}

__global__ void k_resout(const float* __restrict__ d0ord, const float* __restrict__ h6,
                         float* __restrict__ res)
{
  long long id = (long long)blockIdx.x * 256 + threadIdx.x;
  if (id >= (long long)Bv * 2 * Mv * 6) return;
  const int c = (int)(id % 6);
  long long t = id / 6;
  const int j = (int)(t % (2 * Mv));
  const int b = (int)(t / (2 * Mv));
  res[id] = (j < Mv) ? d0ord[((long long)b * Mv + j) * 6 + c]
                     : h6[((long long)b * 6 + c) * (long long)Mv + (j - Mv)];
}

// ---------------------------------------------------------------------------
extern "C" void kernel_launch(void* const* d_in, const int* in_sizes, int n_in,
                              void* d_out, int out_size, void* d_ws, size_t ws_size,
                              hipStream_t stream)
{
  (void)in_sizes; (void)n_in; (void)out_size; (void)ws_size;
  // Input order: xyz, rand_grid, then params flattened with sorted dict keys:
  // dec[0..3], enc[0..3]{wp,wv}, lin_b, lin_w, mlp[0..2], res[0..5]{wp,wv}
  const float* xyz = (const float*)d_in[0];
  const float* rg  = (const float*)d_in[1];
  const float* decW[4]  = {(const float*)d_in[2], (const float*)d_in[3], (const float*)d_in[4], (const float*)d_in[5]};
  const float* encWp[4] = {(const float*)d_in[6], (const float*)d_in[8], (const float*)d_in[10], (const float*)d_in[12]};
  const float* encWv[4] = {(const float*)d_in[7], (const float*)d_in[9], (const float*)d_in[11], (const float*)d_in[13]};
  const float* lin_b = (const float*)d_in[14];
  const float* lin_w = (const float*)d_in[15];
  const float* mlpW[3] = {(const float*)d_in[16], (const float*)d_in[17], (const float*)d_in[18]};
  const float* resWp[6]; const float* resWv[6];
  for (int l = 0; l < 6; ++l) { resWp[l] = (const float*)d_in[19 + 2 * l]; resWv[l] = (const float*)d_in[20 + 2 * l]; }

  float* out = (float*)d_out;
  float* outCoarse = out;
  float* outDense  = out + (size_t)Bv * NCOARSE * 6;
  float* outLoss   = outDense + (size_t)Bv * NDENSE * 6;
  float* outRes    = outLoss + 1;

  size_t cur = 0;
  auto alloc = [&](size_t nf) -> float* {
    float* p = (float*)d_ws + cur;
    cur += (nf + 63) & ~(size_t)63;
    return p;
  };
  float* xt     = alloc((size_t)Bv * 6 * Nv);
  float* featsA = alloc((size_t)Bv * LATENTv * Nv);
  float* featsB = alloc((size_t)Bv * LATENTv * Nv);
  float* ptbuf  = alloc((size_t)Bv * LATENTv * Nv);
  float* sums   = alloc((size_t)Bv * 256 * 4096);
  float* Vg     = alloc((size_t)Bv * 512 * 4096);
  float* cnt    = alloc((size_t)Bv * 4096);
  int*   flat   = (int*)alloc((size_t)Bv * Mv);
  float* stats  = alloc(64);
  float* fm_t   = alloc((size_t)LATENTv * Bv);
  float* feat_t = alloc((size_t)LATENTv * Bv);
  float* meanmst= alloc(64);
  int*   order  = (int*)alloc((size_t)Bv * Mv);
  int*   fidx   = (int*)alloc((size_t)Bv * NCOARSE);
  float* d0raw  = alloc((size_t)Bv * Mv * 6);
  float* d0ord  = alloc((size_t)Bv * Mv * 6);
  float* xtres  = alloc((size_t)Bv * 6 * Mv);
  float* hbuf   = alloc((size_t)Bv * 1008 * Mv);
  float* h6     = alloc((size_t)Bv * 6 * Mv);

  auto g1 = [](long long tot) { return dim3((unsigned)((tot + 255) / 256)); };

  // GEMM dispatcher: TDM-staged 16x32 kernel for fully regular shapes, generic else.
  auto gemm = [&](const float* W, long long wsz, int wzd,
                  const float* X, long long xsz, int ldx,
                  float* C, long long csz, int ldc, const float* bias,
                  int O, int I, int Ncols, int act, int Z) {
    if ((O % 16 == 0) && (Ncols % NT == 0) && (I % KT == 0)) {
      dim3 g(Ncols / NT, O / 16, Z);
      k_wmma_gemm_tdm<<<g, 32, 0, stream>>>(W, wsz, wzd, X, xsz, ldx, C, csz, ldc, bias, O, I, Ncols, act);
    } else {
      dim3 g((Ncols + 15) / 16, (O + 15) / 16, Z);
      k_wmma_gemm<<<g, 32, 0, stream>>>(W, wsz, wzd, X, xsz, ldx, C, csz, ldc, bias, O, I, Ncols, act);
    }
  };

  // ------------------------------ encoder -----------------------------------
  k_transpose6<<<g1((long long)Bv * 6 * Nv), 256, 0, stream>>>(xyz, xt, Nv);
  k_stats<<<Bv, 256, 0, stream>>>(xt, (long long)6 * Nv, Nv, stats);

  struct LC { int I, O, r; };
  const LC enc[4] = {{6, 64, 16}, {64, 128, 16}, {128, 512, 8}, {512, 1024, 4}};
  const float* fin = xt; long long fzs = (long long)6 * Nv; int ldf = Nv;
  float* cures[2] = {featsA, featsB};
  for (int l = 0; l < 4; ++l) {
    const int I = enc[l].I, O = enc[l].O, r = enc[l].r, r3 = r * r * r;
    const int rs = (r == 4) ? 2 : ((r == 8) ? 3 : 4);
    k_flat<<<g1((long long)Bv * Nv), 256, 0, stream>>>(xt, (long long)6 * Nv, Nv, stats, r, flat);
    hipMemsetAsync(sums, 0, (size_t)Bv * I * r3 * sizeof(float), stream);
    hipMemsetAsync(cnt, 0, (size_t)Bv * r3 * sizeof(float), stream);
    k_scatter<<<g1((long long)Bv * Nv), 256, 0, stream>>>(fin, fzs, ldf, I, flat, Nv, r3, sums, cnt);
    k_avg<<<g1((long long)Bv * I * r3), 256, 0, stream>>>(sums, cnt, I, r3);
    dim3 gv((r3 + 15) / 16, (O + 15) / 16, Bv);
    k_wmma_voxconv<<<gv, 32, 0, stream>>>(encWv[l], sums, Vg, O, I, r, rs);
    gemm(encWp[l], 0, 1, fin, fzs, ldf, ptbuf, (long long)O * Nv, Nv, nullptr, O, I, Nv, 1, Bv);
    float* fo = cures[l & 1];
    k_gather_add<<<g1((long long)Bv * O * Nv), 256, 0, stream>>>(ptbuf, Vg, flat, fo,
                                                                 (long long)O * Nv, Nv, O, Nv, r3);
    fin = fo; fzs = (long long)O * Nv; ldf = Nv;
  }
  k_maxpool<<<g1((long long)LATENTv * Bv), 256, 0, stream>>>(featsB, fm_t);
  gemm(lin_w, 0, 1, fm_t, 0, Bv, feat_t, 0, Bv, lin_b, LATENTv, LATENTv, Bv, 3, 1);

  // ------------------------------ decoder -----------------------------------
  float* yinA = featsA; float* yinB = featsB;
  k_build_yin<<<g1((long long)Pv * Bv * LDv * NPPv), 256, 0, stream>>>(rg, feat_t, yinA);
  const int decI[4] = {1026, 1026, 513, 256};
  const int decO[4] = {1026, 513, 256, 6};
  const float* din = yinA; float* dob = yinB;
  for (int l = 0; l < 4; ++l) {
    const int I = decI[l], O = decO[l];
    gemm(decW[l], (long long)O * I, Bv, din, (long long)I * NPPv, NPPv,
         dob, (long long)O * NPPv, NPPv, nullptr, O, I, NPPv, 3, Pv * Bv);
    float* t = (float*)din; din = dob; dob = t;
  }
  k_coarse<<<g1((long long)Bv * NCOARSE * 6), 256, 0, stream>>>(din, outCoarse);

  // --------------------- expansion penalty / fps / mds ----------------------
  hipMemsetAsync(meanmst, 0, Bv * sizeof(float), stream);
  hipMemsetAsync(outLoss, 0, sizeof(float), stream);
  k_mst<<<Bv * Pv, NPPv, 0, stream>>>(outCoarse, meanmst, outLoss);
  k_fps<<<Bv, 256, 0, stream>>>(outCoarse, fidx);
  k_dense_gather<<<g1((long long)Bv * NDENSE * 6), 256, 0, stream>>>(outCoarse, fidx, outDense);
  k_dense0raw<<<g1((long long)Bv * Mv * 6), 256, 0, stream>>>(xyz, outCoarse, d0raw);
  k_mds<<<Bv, 256, 0, stream>>>(d0raw, meanmst, order);
  k_reorder<<<g1((long long)Bv * Mv * 6), 256, 0, stream>>>(d0raw, order, d0ord);

  // --------------------------- residual network -----------------------------
  k_transpose6<<<g1((long long)Bv * 6 * Mv), 256, 0, stream>>>(d0ord, xtres, Mv);
  k_stats<<<Bv, 256, 0, stream>>>(xtres, (long long)6 * Mv, Mv, stats);

  const LC res[6] = {{6, 16, 4}, {16, 32, 4}, {32, 64, 8}, {64, 128, 8}, {128, 256, 16}, {256, 512, 16}};
  const int choff[6] = {0, 16, 48, 112, 240, 496};
  const float* rin = xtres; long long rzs = (long long)6 * Mv; int rld = Mv;
  for (int l = 0; l < 6; ++l) {
    const int I = res[l].I, O = res[l].O, r = res[l].r, r3 = r * r * r;
    const int rs = (r == 4) ? 2 : ((r == 8) ? 3 : 4);
    k_flat<<<g1((long long)Bv * Mv), 256, 0, stream>>>(xtres, (long long)6 * Mv, Mv, stats, r, flat);
    hipMemsetAsync(sums, 0, (size_t)Bv * I * r3 * sizeof(float), stream);
    hipMemsetAsync(cnt, 0, (size_t)Bv * r3 * sizeof(float), stream);
    k_scatter<<<g1((long long)Bv * Mv), 256, 0, stream>>>(rin, rzs, rld, I, flat, Mv, r3, sums, cnt);
    k_avg<<<g1((long long)Bv * I * r3), 256, 0, stream>>>(sums, cnt, I, r3);
    dim3 gv((r3 + 15) / 16, (O + 15) / 16, Bv);
    k_wmma_voxconv<<<gv, 32, 0, stream>>>(resWv[l], sums, Vg, O, I, r, rs);
    gemm(resWp[l], 0, 1, rin, rzs, rld, ptbuf, (long long)O * Mv, Mv, nullptr, O, I, Mv, 1, Bv);
    float* ho = hbuf + (long long)choff[l] * Mv;
    k_gather_add<<<g1((long long)Bv * O * Mv), 256, 0, stream>>>(ptbuf, Vg, flat, ho,
                                                                 (long long)1008 * Mv, Mv, O, Mv, r3);
    rin = ho; rzs = (long long)1008 * Mv; rld = Mv;
  }
  gemm(mlpW[0], 0, 1, hbuf, (long long)1008 * Mv, Mv, featsA, (long long)512 * Mv, Mv,
       nullptr, 512, 1008, Mv, 1, Bv);
  gemm(mlpW[1], 0, 1, featsA, (long long)512 * Mv, Mv, featsB, (long long)256 * Mv, Mv,
       nullptr, 256, 512, Mv, 1, Bv);
  gemm(mlpW[2], 0, 1, featsB, (long long)256 * Mv, Mv, h6, (long long)6 * Mv, Mv,
       nullptr, 6, 256, Mv, 1, Bv);
  k_resout<<<g1((long long)Bv * 2 * Mv * 6), 256, 0, stream>>>(d0ord, h6, outRes);
}